// NSE_22024592294566
// MI455X (gfx1250) — compile-verified
//
#include <hip/hip_runtime.h>

// Problem sizes (compile-time constants from the reference)
#define V_ 32000
#define E_ 512
#define H_ 512
#define S_ 512
#define B_ 32

typedef __attribute__((ext_vector_type(2))) float v2f;
typedef __attribute__((ext_vector_type(8))) float v8f;

__device__ __forceinline__ float sigmoidf_(float x) { return 1.0f / (1.0f + __expf(-x)); }

// ---------------------------------------------------------------------------
// K1: embedding gather -> emb [S*B, E] (GEMM layout) and M0 [B, S, H] (scan layout)
// ---------------------------------------------------------------------------
__global__ void nse_gather(const int* __restrict__ src, const float* __restrict__ Wemb,
                           float* __restrict__ emb, float* __restrict__ M) {
  const int r = blockIdx.x;                 // r = s*B + b
  const int s = r / B_, b = r % B_;
  const int idx = src[r];
  const float* w  = Wemb + (long long)idx * E_;
  float* er = emb + (long long)r * E_;
  float* mr = M + ((long long)b * S_ + s) * H_;
  for (int e = threadIdx.x; e < E_; e += blockDim.x) {
    const float v = w[e];
    er[e] = v;
    mr[e] = v;
  }
}

// ---------------------------------------------------------------------------
// K2: read-LSTM as fp32 WMMA GEMM with fused activation epilogue.
//   gates = emb[16384,512] x W_ih_r^T  -> only i,g,o gates (f-gate dead: c0=0)
//   o_all[r, h] = sigmoid(o) * tanh(sigmoid(i) * tanh(g))
// One wave32 per 16x16 tile; 3 accumulators share the A fragment.
// ---------------------------------------------------------------------------
__global__ void __launch_bounds__(256) nse_read_lstm(
    const float* __restrict__ emb, const float* __restrict__ Wih,
    const float* __restrict__ bih, const float* __restrict__ bhh,
    float* __restrict__ o_all) {
  const int lane = threadIdx.x & 31;
  const int wave = threadIdx.x >> 5;                 // 8 waves per block
  const int wid  = blockIdx.x * 8 + wave;            // global tile id
  const int rt   = wid >> 5;                         // 1024 row tiles
  const int ct   = wid & 31;                         // 32 col tiles
  const int R    = rt * 16;
  const int Hc   = ct * 16;
  const int hi   = lane >> 4;                        // lane half: K base +0 / +2
  const int l15  = lane & 15;

  const float* arow  = emb + (long long)(R + l15) * E_;
  const float* browI = Wih + (long long)(0 * H_ + Hc + l15) * E_;
  const float* browG = Wih + (long long)(2 * H_ + Hc + l15) * E_;
  const float* browO = Wih + (long long)(3 * H_ + Hc + l15) * E_;

  v8f ai = {}, ag = {}, ao = {};
  const int koff = hi ? 2 : 0;
#pragma unroll 4
  for (int k0 = 0; k0 < E_; k0 += 4) {
    const int kk = k0 + koff;                        // even -> 8B aligned
    const v2f a  = *(const v2f*)(arow + kk);
    const v2f bI = *(const v2f*)(browI + kk);
    const v2f bG = *(const v2f*)(browG + kk);
    const v2f bO = *(const v2f*)(browO + kk);
    ai = __builtin_amdgcn_wmma_f32_16x16x4_f32(false, a, false, bI, (short)0, ai, false, false);
    ag = __builtin_amdgcn_wmma_f32_16x16x4_f32(false, a, false, bG, (short)0, ag, false, false);
    ao = __builtin_amdgcn_wmma_f32_16x16x4_f32(false, a, false, bO, (short)0, ao, false, false);
  }

  const int col = Hc + l15;
  const float bI = bih[0 * H_ + col] + bhh[0 * H_ + col];
  const float bG = bih[2 * H_ + col] + bhh[2 * H_ + col];
  const float bO = bih[3 * H_ + col] + bhh[3 * H_ + col];
#pragma unroll
  for (int v = 0; v < 8; ++v) {
    const int row = R + v + (hi ? 8 : 0);
    const float c = sigmoidf_(ai[v] + bI) * tanhf(ag[v] + bG);
    const float h = sigmoidf_(ao[v] + bO) * tanhf(c);
    o_all[(long long)row * H_ + col] = h;           // o_all is [S*B, H]
  }
}

// ---------------------------------------------------------------------------
// K3: weight fusion (fp32 WMMA GEMM):
//   W_fused[j, k] = sum_p W_ihw[jmap(j), p] * W_c[p, k]   j in [0,1536), k in [0,1024)
// jmap: fused rows 0..511 = i rows 0..511; 512..1023 = g rows 1024..1535;
//       1024..1535 = o rows 1536..2047  ->  jmap = j<512 ? j : j+512
// ---------------------------------------------------------------------------
__global__ void __launch_bounds__(256) nse_fuse_w(
    const float* __restrict__ Wihw, const float* __restrict__ Wc,
    float* __restrict__ Wf) {
  const int lane = threadIdx.x & 31;
  const int wave = threadIdx.x >> 5;
  const int wid  = blockIdx.x * 8 + wave;            // 6144 tiles
  const int rt   = wid >> 6;                         // 96 row tiles
  const int ct   = wid & 63;                         // 64 col tiles
  const int R    = rt * 16;
  const int Cc   = ct * 16;
  const int hi   = lane >> 4;
  const int l15  = lane & 15;

  const int arowj = R + l15;
  const int jmap  = (arowj < H_) ? arowj : arowj + H_;
  const float* arow = Wihw + (long long)jmap * (2 * H_);   // contiguous in p
  const int col = Cc + l15;

  v8f acc = {};
  const int koff = hi ? 2 : 0;
#pragma unroll 4
  for (int k0 = 0; k0 < 2 * H_; k0 += 4) {
    const int kk = k0 + koff;
    const v2f a = *(const v2f*)(arow + kk);
    v2f bv;                                          // B[p, col] = W_c[p, col]
    bv.x = Wc[(long long)kk * (2 * H_) + col];
    bv.y = Wc[(long long)(kk + 1) * (2 * H_) + col];
    acc = __builtin_amdgcn_wmma_f32_16x16x4_f32(false, a, false, bv, (short)0, acc, false, false);
  }
#pragma unroll
  for (int v = 0; v < 8; ++v) {
    const int orow = R + v + (hi ? 8 : 0);
    Wf[(long long)orow * (2 * H_) + col] = acc[v];
  }
}

// b_fused[j] = W_ihw[jmap(j),:] . b_c + b_ihw[jmap] + b_hhw[jmap]
__global__ void __launch_bounds__(512) nse_fuse_b(
    const float* __restrict__ Wihw, const float* __restrict__ bc,
    const float* __restrict__ bihw, const float* __restrict__ bhhw,
    float* __restrict__ bf) {
  const int tid = threadIdx.x;
#pragma unroll
  for (int g = 0; g < 3; ++g) {
    const int j    = g * H_ + tid;
    const int jmap = (j < H_) ? j : j + H_;
    const float* wr = Wihw + (long long)jmap * (2 * H_);
    float p = 0.f;
#pragma unroll 4
    for (int k = 0; k < 2 * H_; ++k) p += wr[k] * bc[k];
    bf[j] = p + bihw[jmap] + bhhw[jmap];
  }
}

// ---------------------------------------------------------------------------
// K4: per-batch sequential scan. One workgroup (16 wave32s) per batch element.
// M[b] (1 MB each, 33.5 MB total) stays L2-resident in global scratch.
// Per step: softmax(prev logits) -> weighted read m -> fused GEMV (1536 dots)
// -> activation/output -> fused {M update + next-step logits} (skipped at t=S-1).
// ---------------------------------------------------------------------------
__global__ void __launch_bounds__(512) nse_scan(
    const float* __restrict__ o_all, float* __restrict__ M,
    const float* __restrict__ Wf, const float* __restrict__ bf,
    float* __restrict__ out) {
  const int b    = blockIdx.x;
  const int tid  = threadIdx.x;           // 0..511
  const int lane = tid & 31;
  const int wave = tid >> 5;              // 16 waves
  float* Mb = M + (long long)b * S_ * H_;

  __shared__ float s_ot[2][H_];           // double-buffered o_t
  __shared__ float s_logit[S_];
  __shared__ float s_z[S_];
  __shared__ float s_x[2 * H_];           // [o_t ; m]
  __shared__ float s_g[3 * H_];           // i, g, o gate pre-activations
  __shared__ float s_h[H_];
  __shared__ float s_red[16];

  // ---- preload o_0 and compute initial logits from M0 ----
  s_ot[0][tid] = o_all[(long long)b * H_ + tid];
  __syncthreads();
  for (int i = 0; i < 32; ++i) {
    const int srow = wave * 32 + i;
    const float* mr = Mb + (long long)srow * H_;
    float p = 0.f;
#pragma unroll 4
    for (int k = lane; k < H_; k += 32) p += mr[k] * s_ot[0][k];
    for (int off = 16; off > 0; off >>= 1) p += __shfl_down(p, off, 32);
    if (lane == 0) s_logit[srow] = p;
  }
  __syncthreads();

  for (int t = 0; t < S_; ++t) {
    const int par = t & 1;

    // ---- softmax over S=512 logits ----
    const float v = s_logit[tid];
    float mx = v;
    for (int off = 16; off > 0; off >>= 1) mx = fmaxf(mx, __shfl_down(mx, off, 32));
    if (lane == 0) s_red[wave] = mx;
    __syncthreads();
    if (tid == 0) {
      float mm = s_red[0];
      for (int i = 1; i < 16; ++i) mm = fmaxf(mm, s_red[i]);
      s_red[0] = mm;
    }
    __syncthreads();
    const float e = __expf(v - s_red[0]);
    __syncthreads();                      // all lanes done reading s_red[0]
    float sm = e;
    for (int off = 16; off > 0; off >>= 1) sm += __shfl_down(sm, off, 32);
    if (lane == 0) s_red[wave] = sm;
    __syncthreads();
    if (tid == 0) {
      float ss = 0.f;
      for (int i = 0; i < 16; ++i) ss += s_red[i];
      s_red[0] = 1.0f / ss;
    }
    __syncthreads();
    s_z[tid] = e * s_red[0];
    __syncthreads();

    // ---- weighted read m (coalesced over h) + build x = [o_t ; m] ----
    {
      float acc = 0.f;
#pragma unroll 4
      for (int s = 0; s < S_; ++s) acc += s_z[s] * Mb[(long long)s * H_ + tid];
      s_x[H_ + tid] = acc;
      s_x[tid] = s_ot[par][tid];
    }
    __syncthreads();

    // ---- fused GEMV: 1536 gate pre-activations, dots of length 1024 ----
    for (int jj = 0; jj < 96; ++jj) {
      const int q = wave * 96 + jj;       // 0..1535
      const float* wr = Wf + (long long)q * (2 * H_);
      if (jj + 1 < 96)                    // gfx1250 global_prefetch_b8 of next row
        __builtin_prefetch(Wf + (long long)(q + 1) * (2 * H_) + lane * 32, 0, 1);
      float p = 0.f;
#pragma unroll 4
      for (int k = lane; k < 2 * H_; k += 32) p += wr[k] * s_x[k];
      for (int off = 16; off > 0; off >>= 1) p += __shfl_down(p, off, 32);
      if (lane == 0) s_g[q] = p + bf[q];
    }
    __syncthreads();

    // ---- activations + outputs ----
    {
      const float cell = sigmoidf_(s_g[tid]) * tanhf(s_g[H_ + tid]);
      const float h    = sigmoidf_(s_g[2 * H_ + tid]) * tanhf(cell);
      s_h[tid] = h;
      out[((long long)t * B_ + b) * H_ + tid] = h;             // outputs [S,B,H]
      if (t == S_ - 1) {
        out[(long long)S_ * B_ * H_ + (long long)b * H_ + tid] = h;                         // h_n
        out[(long long)S_ * B_ * H_ + (long long)B_ * H_ + (long long)b * H_ + tid] = cell; // c_n
      }
    }
    // load o_{t+1} for the fused pass
    if (t + 1 < S_) s_ot[par ^ 1][tid] = o_all[((long long)(t + 1) * B_ + b) * H_ + tid];
    __syncthreads();

    // ---- fused: erase/write M update + next-step attention logits ----
    // (final M is never returned, so the whole pass is skipped at t = S-1)
    if (t + 1 < S_) {
      for (int i = 0; i < 32; ++i) {
        const int srow = wave * 32 + i;
        const float zv  = s_z[srow];
        const float omz = 1.0f - zv;
        float* mr = Mb + (long long)srow * H_;
        float p = 0.f;
#pragma unroll 4
        for (int k = lane; k < H_; k += 32) {
          const float nm = omz * mr[k] + zv * s_h[k];
          mr[k] = nm;
          p += nm * s_ot[par ^ 1][k];
        }
        for (int off = 16; off > 0; off >>= 1) p += __shfl_down(p, off, 32);
        if (lane == 0) s_logit[srow] = p;
      }
    }
    __syncthreads();
  }
}

// ---------------------------------------------------------------------------
extern "C" void kernel_launch(void* const* d_in, const int* in_sizes, int n_in,
                              void* d_out, int out_size, void* d_ws, size_t ws_size,
                              hipStream_t stream) {
  (void)in_sizes; (void)n_in; (void)out_size; (void)ws_size;
  const int*   src  = (const int*)d_in[0];
  const float* Wemb = (const float*)d_in[1];
  const float* Wihr = (const float*)d_in[2];
  const float* bihr = (const float*)d_in[3];
  const float* bhhr = (const float*)d_in[4];
  const float* Wc   = (const float*)d_in[5];
  const float* bc   = (const float*)d_in[6];
  const float* Wihw = (const float*)d_in[7];
  const float* bihw = (const float*)d_in[8];
  const float* bhhw = (const float*)d_in[9];
  float* out = (float*)d_out;

  // Workspace: emb [S*B,E] | M [B,S,H] | o_all [S*B,H] | W_fused [1536,1024] | b_fused [1536]
  float* emb   = (float*)d_ws;
  float* M     = emb   + (size_t)S_ * B_ * E_;
  float* o_all = M     + (size_t)B_ * S_ * H_;
  float* Wf    = o_all + (size_t)S_ * B_ * H_;
  float* bf    = Wf    + (size_t)(3 * H_) * (2 * H_);

  nse_gather<<<S_ * B_, 256, 0, stream>>>(src, Wemb, emb, M);

  // 1024 row-tiles x 32 col-tiles = 32768 wave-tiles, 8 waves/block
  nse_read_lstm<<<32768 / 8, 256, 0, stream>>>(emb, Wihr, bihr, bhhr, o_all);

  // 96 row-tiles x 64 col-tiles = 6144 wave-tiles
  nse_fuse_w<<<6144 / 8, 256, 0, stream>>>(Wihw, Wc, Wf);
  nse_fuse_b<<<1, 512, 0, stream>>>(Wihw, bc, bihw, bhhw, bf);

  nse_scan<<<B_, 512, 0, stream>>>(o_all, M, Wf, bf, out);
}